// CoStGcnBlock_40132174414468
// MI455X (gfx1250) — compile-verified
//
#include <hip/hip_runtime.h>
#include <hip/hip_bf16.h>

// ---------------- problem constants (from reference setup_inputs) -----------
#define N_   64
#define C_   64
#define T_   300
#define V_   25
#define S_   3
#define KT   9          // temporal kernel
#define PAD  4
#define EPSB 1e-5f

// ---------------- tiling -----------------
#define TT        16                    // t-values per block
#define NTT       ((T_ + TT - 1) / TT)  // 19 t-tiles
#define COLS      (TT * V_)             // 400 GEMM columns per block
#define NT_TILES  (COLS / 16)           // 25 (= 8 waves * 3 cols + 1 extra)
#define MT_TILES  (C_ / 16)             // 4
#define XSSTR     401                   // xs row stride (17c mod 64 distinct)
#define XATSTR    69                    // xa^T row stride (bank-safe both ways)
#define WGSTR     68                    // weight row stride
#define HT        (TT + KT - 1)         // 24 t-rows incl. conv halo
#define HCOLS     (HT * V_)             // 600
#define HTSTR     69                    // hb^T row stride
#define CNT       (N_ * T_ * V_)        // 480000 elements per channel (BN)

typedef __attribute__((ext_vector_type(2))) float v2f;
typedef __attribute__((ext_vector_type(8))) float v8f;

__device__ __forceinline__ v8f wmma_f32(v2f a, v2f b, v8f c) {
  // V_WMMA_F32_16X16X4_F32 : D = A(16x4,f32) * B(4x16,f32) + C(16x16,f32)
  return __builtin_amdgcn_wmma_f32_16x16x4_f32(false, a, false, b, (short)0, c,
                                               false, false);
}

__device__ __forceinline__ v2f ld2(const float* p, int ks) {
  v2f r;
  r.x = p[4 * ks + 0];   // -> ds_load_2addr_b32 offset0:4ks offset1:4ks+1
  r.y = p[4 * ks + 1];
  return r;
}

// ---------------------------------------------------------------------------
// Kernel 0: zero the BN accumulators (sum1, sq1, sum2, sq2 : 4*64 floats)
// ---------------------------------------------------------------------------
__global__ __launch_bounds__(256) void k_zero(float* st) {
  st[threadIdx.x] = 0.0f;
}

// ---------------------------------------------------------------------------
// Kernel 1: GCN. h0[n,o,t,w] = sum_s sum_c wg[s,o,c]*(sum_v x[n,c,t,v]*Ae[s,v,w])
//                + sum_s bg[s,o]
// One block per (n, t-tile). GEMM: A = wg[s] (o x c), B^T staged as xa_t[col][c].
// Waves 0..7 own nt columns {wave, wave+8, wave+16}; wave 0 also owns nt=24.
// ---------------------------------------------------------------------------
__global__ __launch_bounds__(256) void k_gcn(const float* __restrict__ x,
                                             const float* __restrict__ A,
                                             const float* __restrict__ gattn,
                                             const float* __restrict__ wg,
                                             const float* __restrict__ bg,
                                             float* __restrict__ h0) {
  extern __shared__ float smem[];
  float* xs  = smem;                     // [64][XSSTR]   x tile (row-major)
  float* xat = xs  + C_ * XSSTR;         // [COLS][XATSTR] xa transposed
  float* wgs = xat + COLS * XATSTR;      // [64][WGSTR]   wg for current s
  float* aes = wgs + C_ * WGSTR;         // [625]         Ae for current s
  float* bgs = aes + V_ * V_;            // [64]          sum_s bg[s][o]

  const int n    = blockIdx.x / NTT;
  const int t0   = (blockIdx.x % NTT) * TT;
  const int tid  = threadIdx.x;
  const int lane = tid & 31;
  const int wave = (tid >> 5) & 7;
  const int wsc  = __builtin_amdgcn_readfirstlane(tid) >> 5;  // scalar wave id
  const int hl   = lane >> 4;            // half-wave (0/1)
  const int ll   = lane & 15;

  // stage x[n, c, t0..t0+15, :] -> xs[c][tt*25+w]  (zero-pad past T)
  for (int idx = tid; idx < C_ * COLS; idx += 256) {
    int c = idx / COLS, col = idx % COLS;
    int tt = col / V_, w = col % V_;
    int t = t0 + tt;
    xs[c * XSSTR + col] =
        (t < T_) ? x[((n * C_ + c) * T_ + t) * V_ + w] : 0.0f;
  }
  if (tid < C_) bgs[tid] = bg[tid] + bg[C_ + tid] + bg[2 * C_ + tid];

  v8f acc[4][MT_TILES];
#pragma unroll
  for (int j = 0; j < 4; ++j)
#pragma unroll
    for (int mt = 0; mt < MT_TILES; ++mt) {
      v8f z = {0.f, 0.f, 0.f, 0.f, 0.f, 0.f, 0.f, 0.f};
      acc[j][mt] = z;
    }

  // loop-invariant fragment base pointers (immediate ds offsets in ks loop)
  const float* apb[MT_TILES];
#pragma unroll
  for (int mt = 0; mt < MT_TILES; ++mt)
    apb[mt] = &wgs[(mt * 16 + ll) * WGSTR + 2 * hl];
  const float* bpb[3];
#pragma unroll
  for (int j = 0; j < 3; ++j)
    bpb[j] = &xat[((wave + 8 * j) * 16 + ll) * XATSTR + 2 * hl];
  const float* bpb3 = &xat[(24 * 16 + ll) * XATSTR + 2 * hl];

  for (int s = 0; s < S_; ++s) {
    __syncthreads();  // previous xat/wgs fully consumed, xs/bgs ready
    for (int idx = tid; idx < V_ * V_; idx += 256)
      aes[idx] = A[s * V_ * V_ + idx] * gattn[s * V_ * V_ + idx];
    for (int idx = tid; idx < C_ * C_; idx += 256) {
      int o = idx >> 6, c = idx & 63;
      wgs[o * WGSTR + c] = wg[(s * C_ + o) * C_ + c];
    }
    __syncthreads();
    // xa_t[col][c] = sum_v xs[c][tt*25+v] * Ae[s][v][w]
    for (int idx = tid; idx < C_ * COLS; idx += 256) {
      int col = idx >> 6, c = idx & 63;   // c fastest: LDS-write coalesced
      int tt = col / V_, w = col % V_;
      const float* xrow = &xs[c * XSSTR + tt * V_];
      float a0 = 0.f;
#pragma unroll
      for (int v = 0; v < V_; ++v) a0 += xrow[v] * aes[v * V_ + w];
      xat[col * XATSTR + c] = a0;
    }
    __syncthreads();
    // WMMA: 4 A-frags + 3(4) B-frags per ks, 12(16) WMMAs
    for (int ks = 0; ks < C_ / 4; ++ks) {
      v2f am[MT_TILES];
#pragma unroll
      for (int mt = 0; mt < MT_TILES; ++mt) am[mt] = ld2(apb[mt], ks);
#pragma unroll
      for (int j = 0; j < 3; ++j) {     // nt = wave+8j <= 23 : no guard
        v2f b = ld2(bpb[j], ks);
#pragma unroll
        for (int mt = 0; mt < MT_TILES; ++mt)
          acc[j][mt] = wmma_f32(am[mt], b, acc[j][mt]);
      }
      if (wsc == 0) {                   // scalar branch: wave 0 owns nt=24
        v2f b = ld2(bpb3, ks);
#pragma unroll
        for (int mt = 0; mt < MT_TILES; ++mt)
          acc[3][mt] = wmma_f32(am[mt], b, acc[3][mt]);
      }
    }
  }

  // epilogue: + sum_s bg[s][o], store h0[n][o][t][w]
#pragma unroll
  for (int j = 0; j < 4; ++j) {
    int nt = (j < 3) ? (wave + 8 * j) : 24;
    if (j == 3 && wsc != 0) continue;
    int col = nt * 16 + ll;
    int tt = col / V_, w = col % V_;
    int t = t0 + tt;
    if (t < T_) {
#pragma unroll
      for (int mt = 0; mt < MT_TILES; ++mt) {
#pragma unroll
        for (int r = 0; r < 8; ++r) {
          int o = mt * 16 + 8 * hl + r;
          h0[((n * C_ + o) * T_ + t) * V_ + w] = acc[j][mt][r] + bgs[o];
        }
      }
    }
  }
}

// ---------------------------------------------------------------------------
// Kernel 2: per-channel sum / sum-of-squares over (N,T,V)  (BN training stats)
// grid = (30, 64); blockIdx.y = channel; thread r covers all n for fixed (t,w)
// ---------------------------------------------------------------------------
__global__ __launch_bounds__(256) void k_stats(const float* __restrict__ buf,
                                               float* __restrict__ sum,
                                               float* __restrict__ sumsq) {
  const int c = blockIdx.y;
  const int tid = threadIdx.x;
  const int r = blockIdx.x * 256 + tid;
  const int per = T_ * V_;  // 7500
  float s = 0.f, q = 0.f;
  if (r < per) {
    for (int nn = 0; nn < N_; ++nn) {
      float v = buf[(nn * C_ + c) * per + r];
      s += v;
      q += v * v;
    }
  }
  __shared__ float rs[256], rq[256];
  rs[tid] = s;
  rq[tid] = q;
  __syncthreads();
  for (int st = 128; st > 0; st >>= 1) {
    if (tid < st) { rs[tid] += rs[tid + st]; rq[tid] += rq[tid + st]; }
    __syncthreads();
  }
  if (tid == 0) {
    atomicAdd(&sum[c], rs[0]);
    atomicAdd(&sumsq[c], rq[0]);
  }
}

// ---------------------------------------------------------------------------
// Kernel 3: temporal conv.  hbn = relu(bn1(h0) + x)  (staged transposed in LDS,
//           zero-padded in t).  y[n,o,t,w] = sum_k sum_c hbn[n,c,t+k-4,w]*wt[o,c,k] + bt[o]
// ---------------------------------------------------------------------------
__global__ __launch_bounds__(256) void k_tconv(const float* __restrict__ h0,
                                               const float* __restrict__ x,
                                               const float* __restrict__ wt,
                                               const float* __restrict__ bt,
                                               const float* __restrict__ gamma1,
                                               const float* __restrict__ beta1,
                                               const float* __restrict__ sum1,
                                               const float* __restrict__ sq1,
                                               float* __restrict__ y) {
  extern __shared__ float smem[];
  float* hbt = smem;                     // [HCOLS][HTSTR] hbn transposed
  float* wks = hbt + HCOLS * HTSTR;      // [64][WGSTR]    wt[:, :, k]
  float* scs = wks + C_ * WGSTR;         // [64] BN1 scale
  float* shs = scs + C_;                 // [64] BN1 shift
  float* bts = shs + C_;                 // [64] bt

  const int n    = blockIdx.x / NTT;
  const int t0   = (blockIdx.x % NTT) * TT;
  const int tid  = threadIdx.x;
  const int lane = tid & 31;
  const int wave = (tid >> 5) & 7;
  const int wsc  = __builtin_amdgcn_readfirstlane(tid) >> 5;
  const int hl   = lane >> 4;
  const int ll   = lane & 15;

  if (tid < C_) {
    float mean = sum1[tid] * (1.0f / CNT);
    float var  = sq1[tid] * (1.0f / CNT) - mean * mean;
    float sc   = gamma1[tid] * rsqrtf(var + EPSB);
    scs[tid] = sc;
    shs[tid] = beta1[tid] - mean * sc;
    bts[tid] = bt[tid];
  }
  __syncthreads();

  // stage hbn for t in [t0-4, t0+19]; zero outside [0,T) (conv zero padding)
  for (int idx = tid; idx < C_ * HCOLS; idx += 256) {
    int c = idx / HCOLS, j = idx % HCOLS;  // j fastest: global-read coalesced
    int tl = j / V_, w = j % V_;
    int t = t0 - PAD + tl;
    float v = 0.0f;
    if (t >= 0 && t < T_) {
      int g = ((n * C_ + c) * T_ + t) * V_ + w;
      v = h0[g] * scs[c] + shs[c] + x[g];
      v = v > 0.0f ? v : 0.0f;  // relu (gcn residual already added)
    }
    hbt[j * HTSTR + c] = v;
  }

  v8f acc[4][MT_TILES];
#pragma unroll
  for (int j = 0; j < 4; ++j)
#pragma unroll
    for (int mt = 0; mt < MT_TILES; ++mt) {
      v8f z = {0.f, 0.f, 0.f, 0.f, 0.f, 0.f, 0.f, 0.f};
      acc[j][mt] = z;
    }

  const float* apb[MT_TILES];
#pragma unroll
  for (int mt = 0; mt < MT_TILES; ++mt)
    apb[mt] = &wks[(mt * 16 + ll) * WGSTR + 2 * hl];

  for (int k = 0; k < KT; ++k) {
    __syncthreads();  // previous wks consumed / hbt ready
    for (int idx = tid; idx < C_ * C_; idx += 256) {
      int o = idx >> 6, c = idx & 63;
      wks[o * WGSTR + c] = wt[((o * C_ + c) * KT + k)];
    }
    __syncthreads();
    const float* bpb[3];
#pragma unroll
    for (int j = 0; j < 3; ++j)
      bpb[j] = &hbt[((wave + 8 * j) * 16 + ll + k * V_) * HTSTR + 2 * hl];
    const float* bpb3 = &hbt[(24 * 16 + ll + k * V_) * HTSTR + 2 * hl];

    for (int ks = 0; ks < C_ / 4; ++ks) {
      v2f am[MT_TILES];
#pragma unroll
      for (int mt = 0; mt < MT_TILES; ++mt) am[mt] = ld2(apb[mt], ks);
#pragma unroll
      for (int j = 0; j < 3; ++j) {
        v2f b = ld2(bpb[j], ks);
#pragma unroll
        for (int mt = 0; mt < MT_TILES; ++mt)
          acc[j][mt] = wmma_f32(am[mt], b, acc[j][mt]);
      }
      if (wsc == 0) {
        v2f b = ld2(bpb3, ks);
#pragma unroll
        for (int mt = 0; mt < MT_TILES; ++mt)
          acc[3][mt] = wmma_f32(am[mt], b, acc[3][mt]);
      }
    }
  }

  // epilogue: + bt[o], store y
#pragma unroll
  for (int j = 0; j < 4; ++j) {
    int nt = (j < 3) ? (wave + 8 * j) : 24;
    if (j == 3 && wsc != 0) continue;
    int col = nt * 16 + ll;
    int tt = col / V_, w = col % V_;
    int t = t0 + tt;
    if (t < T_) {
#pragma unroll
      for (int mt = 0; mt < MT_TILES; ++mt) {
#pragma unroll
        for (int r = 0; r < 8; ++r) {
          int o = mt * 16 + 8 * hl + r;
          y[((n * C_ + o) * T_ + t) * V_ + w] = acc[j][mt][r] + bts[o];
        }
      }
    }
  }
}

// ---------------------------------------------------------------------------
// Kernel 4: out = relu(bn2(y) + x)
// ---------------------------------------------------------------------------
__global__ __launch_bounds__(256) void k_final(const float* __restrict__ yb,
                                               const float* __restrict__ x,
                                               const float* __restrict__ gamma2,
                                               const float* __restrict__ beta2,
                                               const float* __restrict__ sum2,
                                               const float* __restrict__ sq2,
                                               float* __restrict__ out) {
  __shared__ float scs[C_], shs[C_];
  const int tid = threadIdx.x;
  if (tid < C_) {
    float mean = sum2[tid] * (1.0f / CNT);
    float var  = sq2[tid] * (1.0f / CNT) - mean * mean;
    float sc   = gamma2[tid] * rsqrtf(var + EPSB);
    scs[tid] = sc;
    shs[tid] = beta2[tid] - mean * sc;
  }
  __syncthreads();
  const int per = T_ * V_;
  const int total = N_ * C_ * per;
  for (int idx = blockIdx.x * 256 + tid; idx < total; idx += gridDim.x * 256) {
    int c = (idx / per) & (C_ - 1);
    float v = yb[idx] * scs[c] + shs[c] + x[idx];
    out[idx] = v > 0.0f ? v : 0.0f;
  }
}

// ---------------------------------------------------------------------------
extern "C" void kernel_launch(void* const* d_in, const int* in_sizes, int n_in,
                              void* d_out, int out_size, void* d_ws,
                              size_t ws_size, hipStream_t stream) {
  const float* x      = (const float*)d_in[0];
  const float* A      = (const float*)d_in[1];
  const float* gattn  = (const float*)d_in[2];
  const float* wg     = (const float*)d_in[3];
  const float* bg     = (const float*)d_in[4];
  const float* gamma1 = (const float*)d_in[5];
  const float* beta1  = (const float*)d_in[6];
  const float* wt     = (const float*)d_in[7];
  const float* bt     = (const float*)d_in[8];
  const float* gamma2 = (const float*)d_in[9];
  const float* beta2  = (const float*)d_in[10];
  float* out = (float*)d_out;

  const size_t NCTV = (size_t)N_ * C_ * T_ * V_;  // 30,720,000
  float* h0 = (float*)d_ws;
  float* yb = h0 + NCTV;
  float* st = yb + NCTV;  // sum1[64] sq1[64] sum2[64] sq2[64]

  k_zero<<<1, 256, 0, stream>>>(st);

  const size_t lds1 =
      (size_t)(C_ * XSSTR + COLS * XATSTR + C_ * WGSTR + V_ * V_ + C_ + 16) *
      sizeof(float);
  k_gcn<<<dim3(N_ * NTT), 256, lds1, stream>>>(x, A, gattn, wg, bg, h0);

  k_stats<<<dim3(30, C_), 256, 0, stream>>>(h0, st + 0, st + 64);

  const size_t lds3 =
      (size_t)(HCOLS * HTSTR + C_ * WGSTR + 3 * C_ + 16) * sizeof(float);
  k_tconv<<<dim3(N_ * NTT), 256, lds3, stream>>>(h0, x, wt, bt, gamma1, beta1,
                                                 st + 0, st + 64, yb);

  k_stats<<<dim3(30, C_), 256, 0, stream>>>(yb, st + 128, st + 192);

  k_final<<<dim3(4096), 256, 0, stream>>>(yb, x, gamma2, beta2, st + 128,
                                          st + 192, out);
}